// FCNN_heteroBessel_41798621724977
// MI455X (gfx1250) — compile-verified
//
#include <hip/hip_runtime.h>

// ---------------------------------------------------------------------------
// out = (Σ_h sin(sin_w·θ + sin_b)) * (out_w · [J2(j2_lin); J4(j4_lin)]) + out_b
// Two-kernel deterministic reduction; wave-level sums via V_WMMA_F32_16X16X4_F32.
// Cross-half fold via DS_SWIZZLE_B32 SWAPX16 (single instruction, no LDS mem).
// ---------------------------------------------------------------------------

typedef __attribute__((ext_vector_type(2))) float v2f;
typedef __attribute__((ext_vector_type(8))) float v8f;

#define H_UNITS   8192
#define THREADS   256
#define BLOCKS    64   // 64*256 = 16384 = 2H lanes, one Bessel eval each

// ds_swizzle group-of-32 encoding: offset[14:10]=xor, [9:5]=or, [4:0]=and
// SWAPX16: xor=0x10, and=0x1f  ->  (0x10<<10) | 0x1f = 0x401F
__device__ __forceinline__ float lane_xor16(float v) {
  return __int_as_float(
      __builtin_amdgcn_ds_swizzle(__float_as_int(v), 0x401F));
}

// Full-precision wave32 sum using the fp32 WMMA unit.
// A-vgpr0 = A-vgpr1 = p  =>  A[m,0..1]=p[m], A[m,2..3]=p[m+16]  (ISA 16x4 f32 layout)
// B = ones (layout-invariant), C = 0  =>  D[m,n] = 2*(p[m]+p[m+16])
// Sum the 8 D vgprs, fold lane halves with one ds_swizzle SWAPX16, halve.
__device__ __forceinline__ float wave_sum32(float p) {
  v2f a; a[0] = p;    a[1] = p;
  v2f b; b[0] = 1.0f; b[1] = 1.0f;
  v8f c = {};
  c = __builtin_amdgcn_wmma_f32_16x16x4_f32(
      /*neg_a=*/false, a, /*neg_b=*/false, b,
      /*c_mod=*/(short)0, c, /*reuse_a=*/false, /*reuse_b=*/false);
  float s = ((c[0] + c[1]) + (c[2] + c[3])) + ((c[4] + c[5]) + (c[6] + c[7]));
  s += lane_xor16(s);
  return 0.5f * s;
}

// Miller backward recurrence, mirrors jax bessel_jn(z, v=4, n_iter=60).
// J_{k-1} = (2k/z) J_k - J_{k+1}; normalize with 1 = J0 + 2*(J2+J4+...).
// NU is even (2 or 4) so J_NU(-x) = J_NU(x): work on |x|.
template <int NU>
__device__ __forceinline__ float besselJ(float x) {
  const float ax  = __builtin_fabsf(x);
  const float inv = 1.0f / fmaxf(ax, 1e-20f);
  float jp  = 0.0f;      // J_{k+1}
  float jc  = 1.0e-30f;  // J_k seed at k = 60
  float sum = 0.0f;
  float jnu = 0.0f;
#pragma unroll
  for (int k = 60; k >= 1; --k) {
    const float jm = __builtin_fmaf((2.0f * (float)k) * inv, jc, -jp);
    jp = jc;
    jc = jm;
    const int km1 = k - 1;
    if (km1 == NU) jnu = jc;                          // folds at compile time
    if (km1 > 0 && (km1 & 1) == 0) sum += 2.0f * jc;  // even orders
  }
  sum += jc;  // + J0
  const float res = jnu / sum;
  return (ax < 1e-12f) ? 0.0f : res;  // J2(0)=J4(0)=0
}

__global__ __launch_bounds__(THREADS) void fcnn_partial(
    const float* __restrict__ t,
    const float* __restrict__ sin_w, const float* __restrict__ sin_b,
    const float* __restrict__ j2_w,  const float* __restrict__ j2_b,
    const float* __restrict__ j4_w,  const float* __restrict__ j4_b,
    const float* __restrict__ out_w,
    float* __restrict__ ws) {
  const float r     = t[0];
  const float theta = t[1];
  const int g = blockIdx.x * THREADS + threadIdx.x;

  float pS = 0.0f, pD = 0.0f;
  if (g < H_UNITS) {             // block-uniform branch (boundary at block 32)
    pS = __sinf(__builtin_fmaf(sin_w[g], theta, sin_b[g]));
    const float x = __builtin_fmaf(j2_w[g], r, j2_b[g]);
    pD = besselJ<2>(x) * out_w[g];
  } else {
    const int h = g - H_UNITS;
    const float x = __builtin_fmaf(j4_w[h], r, j4_b[h]);
    pD = besselJ<4>(x) * out_w[g];
  }

  const float wS = wave_sum32(pS);   // EXEC all-ones here
  const float wD = wave_sum32(pD);

  __shared__ float sS[THREADS / 32];
  __shared__ float sD[THREADS / 32];
  const int wave = threadIdx.x >> 5;
  const int lane = threadIdx.x & 31;
  if (lane == 0) { sS[wave] = wS; sD[wave] = wD; }
  __syncthreads();
  if (threadIdx.x == 0) {
    float S = 0.0f, D = 0.0f;
#pragma unroll
    for (int i = 0; i < THREADS / 32; ++i) { S += sS[i]; D += sD[i]; }
    ws[blockIdx.x]          = S;   // S partials: ws[0..63]
    ws[BLOCKS + blockIdx.x] = D;   // D partials: ws[64..127]
  }
}

__global__ __launch_bounds__(32) void fcnn_final(
    const float* __restrict__ ws,
    const float* __restrict__ out_b,
    float* __restrict__ out) {
  const int lane = threadIdx.x;  // one full wave32, EXEC all-ones
  const float pS = ws[lane] + ws[lane + 32];
  const float pD = ws[BLOCKS + lane] + ws[BLOCKS + 32 + lane];
  const float S = wave_sum32(pS);
  const float D = wave_sum32(pD);
  if (lane == 0) out[0] = __builtin_fmaf(S, D, out_b[0]);
}

extern "C" void kernel_launch(void* const* d_in, const int* in_sizes, int n_in,
                              void* d_out, int out_size, void* d_ws, size_t ws_size,
                              hipStream_t stream) {
  const float* t     = (const float*)d_in[0];
  const float* sin_w = (const float*)d_in[1];
  const float* sin_b = (const float*)d_in[2];
  const float* j2_w  = (const float*)d_in[3];
  const float* j2_b  = (const float*)d_in[4];
  const float* j4_w  = (const float*)d_in[5];
  const float* j4_b  = (const float*)d_in[6];
  const float* out_w = (const float*)d_in[7];
  const float* out_b = (const float*)d_in[8];
  float* ws  = (float*)d_ws;
  float* out = (float*)d_out;

  fcnn_partial<<<BLOCKS, THREADS, 0, stream>>>(t, sin_w, sin_b, j2_w, j2_b,
                                               j4_w, j4_b, out_w, ws);
  fcnn_final<<<1, 32, 0, stream>>>(ws, out_b, out);
}